// HungarianMatcher_37641093382676
// MI455X (gfx1250) — compile-verified
//
#include <hip/hip_runtime.h>
#include <cmath>

// Problem constants (match reference)
#define NQ    100
#define BSZ   8
#define NC    4
#define NPTS  30
#define NTGT  512
#define P_TOT (NQ * BSZ)        // 800 preds
#define TILE_P 16
#define TILE_T 16
#define POLY_F (NPTS * 2)       // 60 floats per polyline
#define PRED_F (TILE_P * POLY_F) // 960 floats = 240 float4
#define TGT_F  (TILE_T * POLY_F) // 960 floats = 240 float4

#define HAS_ASYNC_LDS __has_builtin(__builtin_amdgcn_global_load_async_to_lds_b128)

#if HAS_ASYNC_LDS
// Match the builtin's prototype exactly: v4i (gcc vector) pointers in AS(1)/AS(3).
typedef int v4i_t __attribute__((vector_size(16)));
typedef v4i_t __attribute__((address_space(1)))* g_v4i_p;
typedef v4i_t __attribute__((address_space(3)))* l_v4i_p;

__device__ __forceinline__ void async_copy16(const float* gsrc, float* lds_dst) {
    float* g_nc = const_cast<float*>(gsrc);
    __builtin_amdgcn_global_load_async_to_lds_b128(
        (g_v4i_p)g_nc, (l_v4i_p)lds_dst, /*offset=*/0, /*cpol=*/0);
}
#endif

__global__ __launch_bounds__(256) void hungarian_cost_kernel(
    const float* __restrict__ logits,     // [800, 4]   (nq,bs flattened)
    const float* __restrict__ pred_poly,  // [800, 60]
    const int*   __restrict__ tgt_ids,    // [512]
    const float* __restrict__ tgt_poly,   // [512, 60]
    float*       __restrict__ out)        // [800, 512]
{
    __shared__ float s_pred[PRED_F];         // 16 pred polylines
    __shared__ float s_tgt[TGT_F];           // 16 target polylines
    __shared__ float s_prob[TILE_P * NC];    // softmax probs per pred
    __shared__ int   s_cls[TILE_T];          // class id per target

    const int tid = threadIdx.x;
    const int p0  = blockIdx.x * TILE_P;
    const int t0  = blockIdx.y * TILE_T;

    // Both tiles are contiguous 3840-byte chunks in global memory.
    const float* gp = pred_poly + (size_t)p0 * POLY_F;
    const float* gt = tgt_poly  + (size_t)t0 * POLY_F;

    // ---- Stage polyline tiles into LDS ------------------------------------
#if HAS_ASYNC_LDS
    {
        int idx = tid;                     // 480 float4 total, 256 threads
        #pragma unroll
        for (int r = 0; r < 2; ++r, idx += 256) {
            if (idx < 240) {
                async_copy16(gp + idx * 4, &s_pred[idx * 4]);
            } else if (idx < 480) {
                const int k = idx - 240;
                async_copy16(gt + k * 4, &s_tgt[k * 4]);
            }
        }
    }
#else
    {
        int idx = tid;
        #pragma unroll
        for (int r = 0; r < 2; ++r, idx += 256) {
            if (idx < 240) {
                ((float4*)s_pred)[idx] = ((const float4*)gp)[idx];
            } else if (idx < 480) {
                const int k = idx - 240;
                ((float4*)s_tgt)[k] = ((const float4*)gt)[k];
            }
        }
    }
#endif

    // ---- Per-pred softmax and per-target class ids (overlaps async copy) --
    if (tid < TILE_P) {
        const float* l = logits + (size_t)(p0 + tid) * NC;
        const float l0 = l[0], l1 = l[1], l2 = l[2], l3 = l[3];
        const float m  = fmaxf(fmaxf(l0, l1), fmaxf(l2, l3));
        const float e0 = expf(l0 - m), e1 = expf(l1 - m);
        const float e2 = expf(l2 - m), e3 = expf(l3 - m);
        const float inv = 1.0f / (e0 + e1 + e2 + e3);
        s_prob[tid * NC + 0] = e0 * inv;
        s_prob[tid * NC + 1] = e1 * inv;
        s_prob[tid * NC + 2] = e2 * inv;
        s_prob[tid * NC + 3] = e3 * inv;
    } else if (tid < TILE_P + TILE_T) {
        const int k = tid - TILE_P;
        s_cls[k] = tgt_ids[t0 + k];
    }

#if HAS_ASYNC_LDS
  #if __has_builtin(__builtin_amdgcn_s_wait_asynccnt)
    __builtin_amdgcn_s_wait_asynccnt(0);
  #else
    asm volatile("s_wait_asynccnt 0" ::: "memory");
  #endif
#endif
    __syncthreads();

    // ---- Chamfer: one (pred, tgt) pair per thread -------------------------
    const int lp = tid & (TILE_P - 1);
    const int lt = tid >> 4;

    // Pred points into registers (broadcast reads: lanes L and L+16 share lp).
    float px[NPTS], py[NPTS];
    #pragma unroll
    for (int i = 0; i < NPTS; ++i) {
        px[i] = s_pred[lp * POLY_F + i * 2 + 0];
        py[i] = s_pred[lp * POLY_F + i * 2 + 1];
    }

    float rmin[NPTS];
    #pragma unroll
    for (int i = 0; i < NPTS; ++i) rmin[i] = 3.0e38f;

    const float* tg = &s_tgt[lt * POLY_F];
    float sum2 = 0.0f;                    // sum over j of min_i D[i][j]
    #pragma unroll 2
    for (int j = 0; j < NPTS; ++j) {
        const float tx = tg[j * 2 + 0];
        const float ty = tg[j * 2 + 1];
        float cm = 3.0e38f;
        #pragma unroll
        for (int i = 0; i < NPTS; ++i) {
            // |dx| + |dy| -> single v_add_f32 with |src| modifiers
            const float d = fabsf(px[i] - tx) + fabsf(py[i] - ty);
            cm      = fminf(cm, d);
            rmin[i] = fminf(rmin[i], d);
        }
        sum2 += cm;
    }

    float sum1 = 0.0f;                    // sum over i of min_j D[i][j]
    #pragma unroll
    for (int i = 0; i < NPTS; ++i) sum1 += rmin[i];

    const float cost_poly = 0.5f * (sum1 + sum2) * (1.0f / (float)NPTS);
    const float cost_cls  = -s_prob[lp * NC + s_cls[lt]];

    out[(size_t)(p0 + lp) * NTGT + (t0 + lt)] = cost_cls + cost_poly;
}

extern "C" void kernel_launch(void* const* d_in, const int* in_sizes, int n_in,
                              void* d_out, int out_size, void* d_ws, size_t ws_size,
                              hipStream_t stream) {
    (void)in_sizes; (void)n_in; (void)out_size; (void)d_ws; (void)ws_size;
    const float* logits    = (const float*)d_in[0];  // [100,8,4]
    const float* pred_poly = (const float*)d_in[1];  // [100,8,30,2]
    const int*   tgt_ids   = (const int*)d_in[2];    // [512]
    const float* tgt_poly  = (const float*)d_in[3];  // [512,30,2]
    float*       out       = (float*)d_out;          // [8,100,512] == flat [800,512]

    dim3 grid(P_TOT / TILE_P, NTGT / TILE_T);        // 50 x 32
    hungarian_cost_kernel<<<grid, 256, 0, stream>>>(logits, pred_poly, tgt_ids,
                                                    tgt_poly, out);
}